// TensorProductRescale_21603685499500
// MI455X (gfx1250) — compile-verified
//
#include <hip/hip_runtime.h>

// ---------------------------------------------------------------------------
// e3nn 'uvu' TensorProductRescale on MI455X (gfx1250), fp32 WMMA formulation.
//
// Per row:  OUT[64x9] = XW[64x51] @ B[51x9]   (+ bias on k==0 column)
//   XW[u,q] = weight[path(q), u] * x[row, xcol(q,u)]
//   B[q,k]  = sum_j W3J_path(q)[iloc(q), j, k] * y[row, j]
// Computed with V_WMMA_F32_16X16X4_F32 (full fp32 precision), 13 K-steps,
// one wave per (row, 16-u tile).  Wigner-3j table built on device in d_ws.
// ---------------------------------------------------------------------------

typedef float v2f __attribute__((ext_vector_type(2)));
typedef float v8f __attribute__((ext_vector_type(8)));

#define NROW_X 576     // floats per x/out row  (64*1 + 64*3 + 64*5)
#define NQ     51      // number of K-channels
#define NQP    52      // padded K (multiple of 4)
#define NPATH  15
#define MULW   64

// path -> (l1, l2, l3)
__device__ __constant__ int P_I1[NPATH] = {0,0,0,1,1,1,1,1,1,2,2,2,2,2,2};
__device__ __constant__ int P_I2[NPATH] = {0,1,2,0,1,1,1,2,2,0,1,1,2,2,2};
__device__ __constant__ int P_I3[NPATH] = {0,1,2,1,0,1,2,1,2,2,1,2,0,1,2};

// q-channel enumeration: q 0..2 -> paths 0..2 (l1=0); q 3..20 -> paths 3..8
// (l1=1, 3 components each); q 21..50 -> paths 9..14 (l1=2, 5 comps each).
__device__ inline int q_path(int q) { return q < 3 ? q : (q < 21 ? 3 + (q - 3) / 3 : 9 + (q - 21) / 5); }
__device__ inline int l_base(int l) { return l == 0 ? 0 : (l == 1 ? 1 : 4); }   // offset of l-block in 9-vec

// per-q A-side addressing: x column = xo + u*xs ; weight base = wo (= path*64)
__device__ inline void qparams(int q, int& xo, int& xs, int& wo) {
  if (q < 3)       { xo = 0;                                  xs = 1; wo = q * 64; }
  else if (q < 21) { int t = q - 3;  int g = t / 3; xo = 64  + (t - 3 * g); xs = 3; wo = (3 + g) * 64; }
  else             { int t = q - 21; int g = t / 5; xo = 256 + (t - 5 * g); xs = 5; wo = (9 + g) * 64; }
}

// ------------------------- Wigner-3j init kernel ---------------------------

__device__ double dfact(int n) { double r = 1.0; for (int i = 2; i <= n; ++i) r *= (double)i; return r; }

__device__ double su2_cg(int j1, int j2, int j3, int m1, int m2, int m3) {
  if (m3 != m1 + m2) return 0.0;
  int vmin = -j1 + j2 + m3;
  if (-j1 + m1 > vmin) vmin = -j1 + m1;
  if (0 > vmin) vmin = 0;
  int vmax = j2 + j3 + m1;
  if (j3 - j1 + j2 < vmax) vmax = j3 - j1 + j2;
  if (j3 + m3 < vmax) vmax = j3 + m3;
  if (vmax < vmin) return 0.0;
  double C = sqrt((2.0 * j3 + 1.0) * dfact(j3 + j1 - j2) * dfact(j3 - j1 + j2) * dfact(j1 + j2 - j3)
                  * dfact(j3 + m3) * dfact(j3 - m3)
                  / (dfact(j1 + j2 + j3 + 1) * dfact(j1 - m1) * dfact(j1 + m1)
                     * dfact(j2 - m2) * dfact(j2 + m2)));
  double S = 0.0;
  for (int v = vmin; v <= vmax; ++v) {
    double t = dfact(j2 + j3 + m1 - v) * dfact(j1 - m1 + v)
             / (dfact(v) * dfact(j3 - j1 + j2 - v) * dfact(j3 + m3 - v) * dfact(v + j1 - j2 - m3));
    S += ((v + j2 + m2) & 1) ? -t : t;
  }
  return C * S;
}

// real->complex SH change-of-basis (e3nn convention), times (-i)^l
__device__ void build_q(int l, double qr[5][5], double qi[5][5]) {
  for (int i = 0; i < 5; ++i)
    for (int j = 0; j < 5; ++j) { qr[i][j] = 0.0; qi[i][j] = 0.0; }
  double s = sqrt(0.5);
  for (int m = -l; m < 0; ++m) {
    qr[l + m][l - m] = s;        // 1/sqrt2 at col l+|m|
    qi[l + m][l + m] = -s;       // -i/sqrt2 at col l-|m|
  }
  qr[l][l] = 1.0;
  for (int m = 1; m <= l; ++m) {
    double sg = (m & 1) ? -1.0 : 1.0;
    qr[l + m][l + m] = sg * s;
    qi[l + m][l - m] = sg * s;   // i*(-1)^m/sqrt2
  }
  if (l == 1) {                  // multiply by -i
    for (int i = 0; i < 5; ++i)
      for (int j = 0; j < 5; ++j) { double r = qr[i][j], im = qi[i][j]; qr[i][j] = im; qi[i][j] = -r; }
  } else if (l == 2) {           // multiply by -1
    for (int i = 0; i < 5; ++i)
      for (int j = 0; j < 5; ++j) { qr[i][j] = -qr[i][j]; qi[i][j] = -qi[i][j]; }
  }
}

// T[q][jg][kg]  (jg,kg in 0..8, dense, zero outside path blocks)
__global__ void w3j_init(float* __restrict__ T) {
  int tid = threadIdx.x;
  for (int i = tid; i < NQ * 81; i += (int)blockDim.x) T[i] = 0.0f;
  __syncthreads();
  int e = tid;                     // one nonzero-region element per thread (615 total)
  int q, off = 0;
  bool found = false;
  for (q = 0; q < NQ; ++q) {
    int p = q_path(q);
    int cnt = (2 * P_I2[p] + 1) * (2 * P_I3[p] + 1);
    if (e < off + cnt) { found = true; break; }
    off += cnt;
  }
  if (!found) return;
  int p  = q_path(q);
  int a  = (q < 3) ? 0 : (q < 21 ? (q - 3) % 3 : (q - 21) % 5);   // i_local on l1 axis
  int l1 = P_I1[p], l2 = P_I2[p], l3 = P_I3[p];
  int d3 = 2 * l3 + 1;
  int rel = e - off;
  int jl = rel / d3, ml = rel % d3;
  double q1r[5][5], q1i[5][5], q2r[5][5], q2i[5][5], q3r[5][5], q3i[5][5];
  build_q(l1, q1r, q1i); build_q(l2, q2r, q2i); build_q(l3, q3r, q3i);
  double acc = 0.0;
  for (int i = 0; i <= 2 * l1; ++i)
    for (int k = 0; k <= 2 * l2; ++k)
      for (int n = 0; n <= 2 * l3; ++n) {
        double cg = su2_cg(l1, l2, l3, i - l1, k - l2, n - l3);
        if (cg == 0.0) continue;
        // q1[i][a] * q2[k][jl] * conj(q3[n][ml]) * cg  (real part)
        double tr = q1r[i][a] * q2r[k][jl] - q1i[i][a] * q2i[k][jl];
        double ti = q1r[i][a] * q2i[k][jl] + q1i[i][a] * q2r[k][jl];
        double vr = tr * q3r[n][ml] + ti * q3i[n][ml];
        acc += vr * cg;
      }
  T[q * 81 + (l_base(l2) + jl) * 9 + (l_base(l3) + ml)] = (float)acc;
}

// ------------------------------ main kernel --------------------------------

__global__ __launch_bounds__(256)
void tp_wmma_kernel(const float* __restrict__ x, const float* __restrict__ y,
                    const float* __restrict__ w, const float* __restrict__ bias,
                    const float* __restrict__ T, float* __restrict__ out, int nrows)
{
  __shared__ float xlds[2 * NROW_X];       // two staged x rows
  __shared__ float wlds[NPATH * MULW];     // weights
  __shared__ float blds[2 * NQP * 16];     // per-row B matrix, padded 52x16

  const int tid = threadIdx.x;
  const long long row0 = (long long)blockIdx.x * 2;
  if (row0 >= nrows) return;
  const int nr = (row0 + 2 <= (long long)nrows) ? 2 : 1;

  for (int i = tid; i < nr * NROW_X; i += 256) xlds[i] = x[row0 * NROW_X + i];
  for (int i = tid; i < NPATH * MULW; i += 256) wlds[i] = w[i];
  if (row0 + 2 < (long long)nrows) __builtin_prefetch(x + (row0 + 2) * NROW_X + tid, 0, 0);

  // build B[q,k] = sum_j T[q][j][k] * y[row][j]  (zero outside blocks / q>=NQ)
  for (int i = tid; i < 2 * NQP * 16; i += 256) {
    int rr  = i / (NQP * 16);
    int rem = i - rr * (NQP * 16);
    int q = rem >> 4;
    int k = rem & 15;
    float acc = 0.0f;
    if (q < NQ && rr < nr) {
      int p  = q_path(q);
      int i2 = P_I2[p], i3 = P_I3[p];
      int kb = l_base(i3);
      if (k >= kb && k < kb + 2 * i3 + 1) {
        int jb = l_base(i2);
        const float* yr = y + (row0 + rr) * 9;
        const float* Tq = T + q * 81 + k;
        for (int j = 0; j <= 2 * i2; ++j)
          acc += Tq[(jb + j) * 9] * yr[jb + j];
      }
    }
    blds[i] = acc;
  }
  __syncthreads();

  const int wv = tid >> 5, lane = tid & 31;
  const int rr = wv >> 2;          // row within block
  const int mt = wv & 3;           // 16-u tile
  if (rr < nr) {
    const int half = lane >> 4;    // K-half selector in A/B fragments
    const int l16  = lane & 15;
    const int u    = mt * 16 + l16;
    const float* xr = xlds + rr * NROW_X;
    const float* br = blds + rr * NQP * 16;

    v8f c = {0.f, 0.f, 0.f, 0.f, 0.f, 0.f, 0.f, 0.f};
    #pragma unroll
    for (int s = 0; s < 13; ++s) {
      int q0 = 4 * s + 2 * half;                 // A frag: v0 holds K=q0, v1 holds K=q0+1
      int q1 = q0 + 1; if (q1 > NQ - 1) q1 = NQ - 1;   // clamp; value killed below
      int xo0, xs0, wo0, xo1, xs1, wo1;
      qparams(q0, xo0, xs0, wo0);
      qparams(q1, xo1, xs1, wo1);
      float a0 = xr[xo0 + u * xs0] * wlds[wo0 + u];
      float a1 = xr[xo1 + u * xs1] * wlds[wo1 + u];
      if (s == 12) a1 = half ? 0.0f : a1;        // K=51 pad lane -> 0
      v2f a; a.x = a0; a.y = a1;
      v2f b;                                     // B frag: rows q0 / q0+1, col = l16
      b.x = br[64 * s + 32 * half + l16];
      b.y = br[64 * s + 32 * half + 16 + l16];
      c = __builtin_amdgcn_wmma_f32_16x16x4_f32(false, a, false, b, (short)0, c, false, false);
    }

    // D tile: VGPR r -> (M = r + 8*half, N = l16); scatter to e3nn layout
    const int k = l16;
    float* orow = out + (row0 + rr) * NROW_X;
    if (k < 9) {
      #pragma unroll
      for (int r = 0; r < 8; ++r) {
        int uu = mt * 16 + half * 8 + r;
        float v = c[r];
        if (k == 0)      orow[uu] = v + bias[uu];
        else if (k < 4)  orow[64 + uu * 3 + (k - 1)] = v;
        else             orow[256 + uu * 5 + (k - 4)] = v;
      }
    }
  }
}

// ------------------------------- launcher ----------------------------------

extern "C" void kernel_launch(void* const* d_in, const int* in_sizes, int n_in,
                              void* d_out, int out_size, void* d_ws, size_t ws_size,
                              hipStream_t stream) {
  const float* x    = (const float*)d_in[0];
  const float* y    = (const float*)d_in[1];
  const float* w    = (const float*)d_in[2];
  const float* bias = (const float*)d_in[3];
  float* out = (float*)d_out;
  float* T   = (float*)d_ws;                 // needs 51*81*4 = 16524 bytes
  const int nrows = in_sizes[0] / NROW_X;    // 131072

  w3j_init<<<1, 1024, 0, stream>>>(T);
  const int nblocks = (nrows + 1) / 2;
  tp_wmma_kernel<<<nblocks, 256, 0, stream>>>(x, y, w, bias, T, out, nrows);
}